// GRULayer_58858231824555
// MI455X (gfx1250) — compile-verified
//
#include <hip/hip_runtime.h>
#include <math.h>

// GRU: B=32, T=512, D=512, H=1024. fp32 end-to-end via V_WMMA_F32_16X16X4_F32.
// Per step: gates kernel then candidate kernel (recurrence serializes T).
// LDS double-buffered K-chunks, staged with async global->LDS (ASYNCcnt).

typedef __attribute__((ext_vector_type(2))) float v2f;
typedef __attribute__((ext_vector_type(8))) float v8f;
typedef __attribute__((ext_vector_type(4))) int   v4i_;
typedef __attribute__((address_space(1))) v4i_    g_v4i;   // global int4
typedef __attribute__((address_space(3))) v4i_    l_v4i;   // LDS int4

#define B_    32
#define T_    512
#define D_    512
#define H_    1024
#define KTOT  1536
#define KCH   192                 // K per LDS chunk
#define NCH   8                   // 1536 / 192
#define PITCH 196                 // 196 % 64 == 4 -> conflict-free row access
#define C4N   (KCH / 4)           // 48 float4 per row per chunk

#if __has_builtin(__builtin_amdgcn_global_load_async_to_lds_b128) && \
    __has_builtin(__builtin_amdgcn_s_wait_asynccnt)
#define USE_ASYNC 1
#else
#define USE_ASYNC 0
#endif

__device__ __forceinline__ void stage16B(float* dst, const float* src) {
#if USE_ASYNC
    __builtin_amdgcn_global_load_async_to_lds_b128(
        (g_v4i*)src, (l_v4i*)dst, 0, 0);
#else
    *(float4*)dst = *(const float4*)src;
#endif
}

// Wait so that at most the most recent stage (12 async instrs/wave) is pending.
__device__ __forceinline__ void wait_async_stage() {
#if USE_ASYNC
    __builtin_amdgcn_s_wait_asynccnt(12);
#endif
}
__device__ __forceinline__ void wait_async_all() {
#if USE_ASYNC
    __builtin_amdgcn_s_wait_asynccnt(0);
#endif
}

__device__ __forceinline__ float fast_sigmoid(float x) {
    return __builtin_amdgcn_rcpf(1.0f + __expf(-x));
}
__device__ __forceinline__ float fast_tanh(float x) {
    return 1.0f - 2.0f * __builtin_amdgcn_rcpf(1.0f + __expf(2.0f * x));
}

// ---------------------------------------------------------------------------
// Kernel 1: gates = sigmoid([x_t, h_{t-1}] @ Wg + bg)
//   r cols [0,H):   rh[m,n]     = r * h_prev[m,n]
//   u cols [H,2H):  ubuf[m,n-H] = u
// Grid: 32 blocks x 128 threads; wave w owns N-tile blockIdx*4+w (128 tiles).
// ---------------------------------------------------------------------------
__global__ __launch_bounds__(128) void gru_gates(
    const float* __restrict__ X, const float* __restrict__ Wg,
    const float* __restrict__ bg, const float* __restrict__ out,
    float* __restrict__ rh, float* __restrict__ ubuf, int t)
{
    __shared__ __align__(16) float lds[2][B_ * PITCH];   // 50,176 B

    const int tid  = threadIdx.x;
    const int lane = tid & 31;
    const int wave = tid >> 5;
    const int n0   = (blockIdx.x * 4 + wave) * 16;
    const int hi   = lane >> 4;          // 0 -> K pair {0,1}; 1 -> {2,3}
    const int m0   = lane & 15;
    const int n    = n0 + m0;

    const int srow = tid >> 2;           // staging: 4 threads per row
    const int sc0  = tid & 3;

    auto stage_chunk = [&](int buf, int kbase) {
        #pragma unroll
        for (int q = 0; q < C4N / 4; ++q) {          // 12 float4 per thread
            const int c4  = sc0 + q * 4;
            const int col = kbase + c4 * 4;
            float* dst = &lds[buf][srow * PITCH + c4 * 4];
            if (col < D_) {
                stage16B(dst, X + (srow * T_ + t) * D_ + col);
            } else if (t > 0) {
                stage16B(dst, out + (srow * T_ + (t - 1)) * H_ + (col - D_));
            } else {
                *(float4*)dst = make_float4(0.f, 0.f, 0.f, 0.f);
            }
        }
    };

    v8f acc0 = {};   // rows 0..15
    v8f acc1 = {};   // rows 16..31

    stage_chunk(0, 0);
    for (int ch = 0; ch < NCH; ++ch) {
        const int kbase = ch * KCH;
        if (ch + 1 < NCH) stage_chunk((ch + 1) & 1, kbase + KCH);
        if (ch + 1 < NCH && t > 0) wait_async_stage();  // stage(ch) retired
        else                       wait_async_all();
        __syncthreads();

        const float* A = lds[ch & 1];
        #pragma unroll 4
        for (int kk = 0; kk < KCH; kk += 4) {
            const int ka = kk + hi * 2;
            v2f a0 = *(const v2f*)(&A[m0 * PITCH + ka]);
            v2f a1 = *(const v2f*)(&A[(m0 + 16) * PITCH + ka]);
            const int kb = kbase + ka;
            v2f b;
            b.x = Wg[kb * (2 * H_) + n];
            b.y = Wg[(kb + 1) * (2 * H_) + n];
            acc0 = __builtin_amdgcn_wmma_f32_16x16x4_f32(
                false, a0, false, b, (short)0, acc0, false, false);
            acc1 = __builtin_amdgcn_wmma_f32_16x16x4_f32(
                false, a1, false, b, (short)0, acc1, false, false);
        }
        __syncthreads();   // all waves done reading buf[ch&1]
    }

    const float bias = bg[n];
    const bool  is_r = (n0 < H_);
    #pragma unroll
    for (int j = 0; j < 8; ++j) {
        const int m = j + hi * 8;        // acc0 -> rows m, acc1 -> rows m+16
        float g0 = fast_sigmoid(acc0[j] + bias);
        float g1 = fast_sigmoid(acc1[j] + bias);
        if (is_r) {
            float hp0 = (t > 0) ? out[(m * T_ + (t - 1)) * H_ + n] : 0.f;
            float hp1 = (t > 0) ? out[((m + 16) * T_ + (t - 1)) * H_ + n] : 0.f;
            rh[m * H_ + n]        = g0 * hp0;
            rh[(m + 16) * H_ + n] = g1 * hp1;
        } else {
            ubuf[m * H_ + (n - H_)]        = g0;
            ubuf[(m + 16) * H_ + (n - H_)] = g1;
        }
    }
}

// ---------------------------------------------------------------------------
// Kernel 2: c = tanh([x_t, rh] @ Wc + bc); h_t = u*h_{t-1} + (1-u)*c
// Grid: 16 blocks x 128 threads; wave w owns N-tile blockIdx*4+w (64 tiles).
// ---------------------------------------------------------------------------
__global__ __launch_bounds__(128) void gru_cand(
    const float* __restrict__ X, const float* __restrict__ Wc,
    const float* __restrict__ bc, const float* __restrict__ rh,
    const float* __restrict__ ubuf, float* __restrict__ out, int t)
{
    __shared__ __align__(16) float lds[2][B_ * PITCH];

    const int tid  = threadIdx.x;
    const int lane = tid & 31;
    const int wave = tid >> 5;
    const int n0   = (blockIdx.x * 4 + wave) * 16;
    const int hi   = lane >> 4;
    const int m0   = lane & 15;
    const int n    = n0 + m0;

    const int srow = tid >> 2;
    const int sc0  = tid & 3;

    auto stage_chunk = [&](int buf, int kbase) {
        #pragma unroll
        for (int q = 0; q < C4N / 4; ++q) {
            const int c4  = sc0 + q * 4;
            const int col = kbase + c4 * 4;
            float* dst = &lds[buf][srow * PITCH + c4 * 4];
            if (col < D_) {
                stage16B(dst, X + (srow * T_ + t) * D_ + col);
            } else {
                stage16B(dst, rh + srow * H_ + (col - D_));
            }
        }
    };

    v8f acc0 = {};
    v8f acc1 = {};

    stage_chunk(0, 0);
    for (int ch = 0; ch < NCH; ++ch) {
        const int kbase = ch * KCH;
        if (ch + 1 < NCH) stage_chunk((ch + 1) & 1, kbase + KCH);
        if (ch + 1 < NCH) wait_async_stage();
        else              wait_async_all();
        __syncthreads();

        const float* A = lds[ch & 1];
        #pragma unroll 4
        for (int kk = 0; kk < KCH; kk += 4) {
            const int ka = kk + hi * 2;
            v2f a0 = *(const v2f*)(&A[m0 * PITCH + ka]);
            v2f a1 = *(const v2f*)(&A[(m0 + 16) * PITCH + ka]);
            const int kb = kbase + ka;
            v2f b;
            b.x = Wc[kb * H_ + n];
            b.y = Wc[(kb + 1) * H_ + n];
            acc0 = __builtin_amdgcn_wmma_f32_16x16x4_f32(
                false, a0, false, b, (short)0, acc0, false, false);
            acc1 = __builtin_amdgcn_wmma_f32_16x16x4_f32(
                false, a1, false, b, (short)0, acc1, false, false);
        }
        __syncthreads();
    }

    const float bias = bc[n];
    #pragma unroll
    for (int j = 0; j < 8; ++j) {
        const int m = j + hi * 8;
        {
            float c  = fast_tanh(acc0[j] + bias);
            float u  = ubuf[m * H_ + n];
            float hp = (t > 0) ? out[(m * T_ + (t - 1)) * H_ + n] : 0.f;
            out[(m * T_ + t) * H_ + n] = u * hp + (1.f - u) * c;
        }
        {
            const int m2 = m + 16;
            float c  = fast_tanh(acc1[j] + bias);
            float u  = ubuf[m2 * H_ + n];
            float hp = (t > 0) ? out[(m2 * T_ + (t - 1)) * H_ + n] : 0.f;
            out[(m2 * T_ + t) * H_ + n] = u * hp + (1.f - u) * c;
        }
    }
}

// ---------------------------------------------------------------------------
extern "C" void kernel_launch(void* const* d_in, const int* in_sizes, int n_in,
                              void* d_out, int out_size, void* d_ws, size_t ws_size,
                              hipStream_t stream) {
    const float* X  = (const float*)d_in[0];   // [32,512,512]
    const float* Wg = (const float*)d_in[1];   // [1536,2048]
    const float* bg = (const float*)d_in[2];   // [2048]
    const float* Wc = (const float*)d_in[3];   // [1536,1024]
    const float* bc = (const float*)d_in[4];   // [1024]
    float* out = (float*)d_out;                // [32,512,1024]

    float* rh = (float*)d_ws;                  // [32,1024]
    float* ub = rh + B_ * H_;                  // [32,1024]

    for (int t = 0; t < T_; ++t) {
        gru_gates<<<dim3(32), dim3(128), 0, stream>>>(X, Wg, bg, out, rh, ub, t);
        gru_cand <<<dim3(16), dim3(128), 0, stream>>>(X, Wc, bc, rh, ub, out, t);
    }
}